// CapsuleLayer_25228637896856
// MI455X (gfx1250) — compile-verified
//
#include <hip/hip_runtime.h>
#include <cstdint>

// Problem constants (match reference setup_inputs()).
#define BB   256      // batch
#define NN   1152     // route nodes
#define CIN  8
#define CC   10       // capsules
#define OO   16       // out dim
#define NO   (NN * OO)            // 18432 floats per (c,b) priors slice
#define NO_BYTES (NO * 4)         // 73728 bytes

typedef float v2f __attribute__((ext_vector_type(2)));
typedef float v8f __attribute__((ext_vector_type(8)));
typedef unsigned int u32x4 __attribute__((ext_vector_type(4)));
typedef int i32x4 __attribute__((ext_vector_type(4)));
typedef int i32x8 __attribute__((ext_vector_type(8)));

// ---------------------------------------------------------------------------
// Kernel A: priors_c[b,n,o] = sum_i x[b,n,i] * W[c,n,i,o]
// For a fixed (c,n): [16 x 8] @ [8 x 16] per b-tile -> V_WMMA_F32_16X16X4_F32
// (two K=4 steps). One block per n; 8 waves each handle 2 b-tiles of 16.
// Output layout in d_ws: [b][n][o]  (contiguous 72KB per b for kernel B).
// ---------------------------------------------------------------------------
__global__ __launch_bounds__(256) void caps_priors_wmma(
    const float* __restrict__ x,        // [B][N][CIN]
    const float* __restrict__ Wc,       // [N][CIN][OO] for this c
    float* __restrict__ priors) {       // [B][N][OO]
  const int n    = blockIdx.x;          // 0..1151
  const int wave = threadIdx.x >> 5;    // 0..7
  const int lane = threadIdx.x & 31;
  const int m    = lane & 15;           // row within 16x16 tile / B column idx
  const int hi   = lane >> 4;           // lane group
  const int kb   = hi * 2;              // K base per ISA f32 A/B layout
  const int o    = lane & 15;

  // B fragments (W[c,n] is 8x16 row-major, 128 contiguous floats).
  const float* wp = Wc + (size_t)n * (CIN * OO);
  v2f bf1, bf2;
  bf1.x = wp[(kb + 0) * OO + o];        // K rows 0..3
  bf1.y = wp[(kb + 1) * OO + o];
  bf2.x = wp[(kb + 4) * OO + o];        // K rows 4..7
  bf2.y = wp[(kb + 5) * OO + o];

  for (int t = wave; t < 16; t += 8) {  // b-tiles of 16
    const int b0 = t * 16;
    const float* xp = x + (size_t)(b0 + m) * (NN * CIN) + n * CIN + kb;
    v2f af1, af2;
    af1.x = xp[0];  af1.y = xp[1];      // K = kb, kb+1
    af2.x = xp[4];  af2.y = xp[5];      // K = kb+4, kb+5
    v8f acc = {};
    acc = __builtin_amdgcn_wmma_f32_16x16x4_f32(false, af1, false, bf1,
                                                (short)0, acc, false, false);
    acc = __builtin_amdgcn_wmma_f32_16x16x4_f32(false, af2, false, bf2,
                                                (short)0, acc, false, false);
    // D layout: VGPR j -> row (j + 8*hi), col o.
#pragma unroll
    for (int j = 0; j < 8; ++j) {
      const int row = j + hi * 8;
      priors[(((size_t)(b0 + row)) * NN + n) * OO + o] = acc[j];
    }
  }
}

// ---------------------------------------------------------------------------
// Kernel B: dynamic routing for one (c,b) per workgroup, priors slice (72KB)
// staged into LDS by the Tensor Data Mover, 3 iterations fully in LDS.
// ---------------------------------------------------------------------------
__global__ __launch_bounds__(256) void caps_route(
    const float* __restrict__ priors,   // [B][N][OO] for this c
    float* __restrict__ out,            // [C][B][OO]
    int c) {
  extern __shared__ float smem[];
  float* prior = smem;                  // 18432
  float* alog  = smem + NO;             // 1152  (routing logits a[n])
  float* p     = alog + NN;             // 1152  (exp weights)
  float* red   = p + NN;                // 256   (block reduction scratch)
  float* sv    = red + 256;             // 16    (pre-squash s)
  float* vv    = sv + 16;               // 16    (squashed v)

  const int tid = threadIdx.x;
  const int b   = blockIdx.x;
  const float* gp = priors + (size_t)b * NO;

  // --- Stage priors slice into LDS with the TDM (one wave issues it). -----
  if (tid < 32) {
    const unsigned long long ga = (unsigned long long)(uintptr_t)gp;
    const unsigned int lds_base = (unsigned int)(uintptr_t)prior; // addr[31:0] == LDS offset
    u32x4 g0;
    g0.x = 0x1u;                                    // count=1, user D#
    g0.y = lds_base;                                // lds_addr
    g0.z = (unsigned int)(ga & 0xFFFFFFFFu);        // global_addr[31:0]
    g0.w = (unsigned int)((ga >> 32) & 0x01FFFFFFu) // global_addr[56:32]
           | (2u << 30);                            // type = 2 ("image")
    i32x8 g1;
    g1[0] = (2 << 16);                  // data_size = 4 bytes
    g1[1] = (int)(18432u << 16);        // tensor_dim0[15:0]=18432 in [31:16]
    g1[2] = (int)(1u << 16);            // tensor_dim0 hi = 0; tensor_dim1 = 1
    g1[3] = (int)(18432u << 16);        // tile_dim0 = 18432
    g1[4] = 1;                          // tile_dim1 = 1
    g1[5] = 18432;                      // tensor_dim0_stride
    g1[6] = 0;
    g1[7] = 0;
    i32x4 gz4 = {0, 0, 0, 0};           // 2D tensor: groups 2/3 unused
    i32x8 gz8 = {0, 0, 0, 0, 0, 0, 0, 0};
    __builtin_amdgcn_tensor_load_to_lds(g0, g1, gz4, gz4, gz8, 0);
    __builtin_amdgcn_s_wait_tensorcnt(0);
  }
  for (int nn = tid; nn < NN; nn += 256) alog[nn] = 0.0f;
  __syncthreads();

  // --- 3 routing iterations entirely in LDS. ------------------------------
  for (int iter = 0; iter < 3; ++iter) {
    // softmax over n: max
    float lm = -3.4e38f;
    for (int nn = tid; nn < NN; nn += 256) lm = fmaxf(lm, alog[nn]);
    red[tid] = lm;
    __syncthreads();
    for (int s = 128; s > 0; s >>= 1) {
      if (tid < s) red[tid] = fmaxf(red[tid], red[tid + s]);
      __syncthreads();
    }
    const float mx = red[0];
    __syncthreads();
    // exp + sum
    float ls = 0.0f;
    for (int nn = tid; nn < NN; nn += 256) {
      const float e = __expf(alog[nn] - mx);
      p[nn] = e;
      ls += e;
    }
    red[tid] = ls;
    __syncthreads();
    for (int s = 128; s > 0; s >>= 1) {
      if (tid < s) red[tid] += red[tid + s];
      __syncthreads();
    }
    const float invZ = 1.0f / red[0];
    __syncthreads();

    // s[o] = invZ * sum_n p[n] * prior[n][o]   (16 n-groups x 16 o lanes)
    {
      const int o = tid & 15, g = tid >> 4;
      float acc = 0.0f;
      for (int nn = g; nn < NN; nn += 16) acc += p[nn] * prior[nn * OO + o];
      red[tid] = acc;
    }
    __syncthreads();
    if (tid < 16) {
      float s = 0.0f;
#pragma unroll
      for (int g = 0; g < 16; ++g) s += red[g * 16 + tid];
      sv[tid] = s * invZ;
    }
    __syncthreads();

    // squash
    if (tid == 0) {
      float sn = 0.0f;
#pragma unroll
      for (int o2 = 0; o2 < OO; ++o2) sn += sv[o2] * sv[o2];
      const float scale = (sn / (1.0f + sn)) * rsqrtf(sn + 1e-8f);
#pragma unroll
      for (int o2 = 0; o2 < OO; ++o2) vv[o2] = scale * sv[o2];
    }
    __syncthreads();

    // agreement update: a[n] += sum_o prior[n][o] * v[o]
    if (iter < 2) {
      for (int nn = tid; nn < NN; nn += 256) {
        float d = 0.0f;
#pragma unroll
        for (int k = 0; k < OO; ++k) {
          const int o2 = (k + tid) & 15;  // rotate to spread LDS banks
          d += prior[nn * OO + o2] * vv[o2];
        }
        alog[nn] += d;
      }
      __syncthreads();
    }
  }

  if (tid < 16) out[((size_t)c * BB + b) * OO + tid] = vv[tid];
}

// ---------------------------------------------------------------------------
// Host: per-capsule pipeline keeps the 18.9MB priors slab L2-resident.
// 10 x (GEMM, route) on one stream; graph capture handles the replay.
// ---------------------------------------------------------------------------
extern "C" void kernel_launch(void* const* d_in, const int* in_sizes, int n_in,
                              void* d_out, int out_size, void* d_ws, size_t ws_size,
                              hipStream_t stream) {
  (void)in_sizes; (void)n_in; (void)out_size; (void)ws_size;
  const float* x  = (const float*)d_in[0];               // [B][N][CIN]
  const float* rw = (const float*)d_in[1];               // [C][N][CIN][OO]
  float* out = (float*)d_out;                            // [C][B][1][1][OO]
  float* ws  = (float*)d_ws;                             // [B][N][OO] (one c at a time)

  const size_t smem_bytes = (size_t)(NO + NN + NN + 256 + 16 + 16) * sizeof(float);
  for (int c = 0; c < CC; ++c) {
    caps_priors_wmma<<<dim3(NN), dim3(256), 0, stream>>>(
        x, rw + (size_t)c * NN * CIN * OO, ws);
    caps_route<<<dim3(BB), dim3(256), smem_bytes, stream>>>(ws, out, c);
  }
}